// KMeans_80848464379958
// MI455X (gfx1250) — compile-verified
//
#include <hip/hip_runtime.h>

#define NB 4
#define NC 128
#define MAXN 15
#define NEGV (-1e30f)
#define VALID_THR (-5e29f)

typedef __attribute__((ext_vector_type(2))) float v2f;
typedef __attribute__((ext_vector_type(8))) float v8f;

// -------------------------------------------------------------------------
// init: cent (B,NC,4) <- input (B,NC,3); centB = (2cx,2cy,2cz,-|c|^2);
//       accum zeroed
// -------------------------------------------------------------------------
__global__ void init_kernel(const float* __restrict__ cent_in,
                            float* __restrict__ cent,
                            float* __restrict__ centB,
                            float* __restrict__ accum) {
    int i = blockIdx.x * blockDim.x + threadIdx.x;   // one thread per (b,c)
    if (i < NB * NC) {
        float x = cent_in[i * 3 + 0];
        float y = cent_in[i * 3 + 1];
        float z = cent_in[i * 3 + 2];
        float* c = cent + i * 4;
        c[0] = x; c[1] = y; c[2] = z; c[3] = 0.0f;
        float* cb = centB + i * 4;
        cb[0] = 2.f * x; cb[1] = 2.f * y; cb[2] = 2.f * z;
        cb[3] = -(x * x + y * y + z * z);
        float* a = accum + i * 4;
        a[0] = 0.f; a[1] = 0.f; a[2] = 0.f; a[3] = 0.f;
    }
}

// -------------------------------------------------------------------------
// assign: one wave = 16 points vs all 128 centroids via 8x WMMA f32 16x16x4
//   A row m = (x, y, z, 1)        (lanes 0-15: K0,K1 ; lanes 16-31: K2,K3)
//   B col n = (2cx,2cy,2cz,-|c|^2) prepared in centB, b64-loadable per lane
//   D[m][n] = 2 p.c - |c|^2  (argmax over n == argmax of full sim)
// -------------------------------------------------------------------------
__global__ void assign_kernel(const float* __restrict__ pts,    // (B,npts,3)
                              const float* __restrict__ centB,  // (B,NC,4) prepared
                              int*  __restrict__ closest,       // (B,npts)
                              float* __restrict__ accum,        // (B,NC,4) x,y,z,count
                              int npts) {
    const int lane = threadIdx.x & 31;
    const int wave = threadIdx.x >> 5;
    const int b    = blockIdx.y;
    const int base = blockIdx.x * 128 + wave * 16;   // 8 waves * 16 points
    const int sub  = lane & 15;
    const int half = lane >> 4;

    // ---- A tile: branch-free via clamped pointer + selects ----
    const int  p     = base + sub;
    const bool valid = (p < npts);
    const int  pc    = valid ? p : 0;
    const float* pp  = pts + ((long long)b * npts + pc) * 3 + half * 2;
    // half==0 -> (x,y) ; half==1 -> (z, 1.0)
    float a0 = valid ? pp[0] : 0.0f;
    float a1 = half ? 1.0f : (valid ? pp[1] : 0.0f);
    v2f a; a.x = a0; a.y = a1;

    float bestv[8];
    int   bestt[8];
#pragma unroll
    for (int v = 0; v < 8; ++v) { bestv[v] = NEGV; bestt[v] = 0; }

    // per-lane B pointer: column n = sub, K-pair selected by half (8B aligned)
    const float* cbp = centB + ((long long)b * NC + sub) * 4 + half * 2;
#pragma unroll
    for (int t = 0; t < 8; ++t) {
        v2f bm = *(const v2f*)(cbp + t * 16 * 4);
        v8f cz8 = {};
        v8f d = __builtin_amdgcn_wmma_f32_16x16x4_f32(
            false, a, false, bm, (short)0, cz8, false, false);
#pragma unroll
        for (int v = 0; v < 8; ++v) {
            float dv = d[v];
            if (dv > bestv[v]) { bestv[v] = dv; bestt[v] = t; }
        }
    }

    // materialize centroid indices, butterfly-argmax within each 16-lane half
    int bi[8];
#pragma unroll
    for (int v = 0; v < 8; ++v) bi[v] = bestt[v] * 16 + sub;
#pragma unroll
    for (int v = 0; v < 8; ++v) {
#pragma unroll
        for (int off = 8; off >= 1; off >>= 1) {
            float ov = __shfl_xor(bestv[v], off, 16);
            int   oi = __shfl_xor(bi[v],   off, 16);
            if (ov > bestv[v]) { bestv[v] = ov; bi[v] = oi; }
        }
    }

    // row M = v + 8*half ; one designated lane (sub==v) per half stores + accumulates
#pragma unroll
    for (int v = 0; v < 8; ++v) {
        if (sub == v) {
            int m  = v + 8 * half;
            int pm = base + m;
            if (pm < npts) {
                int cidx = bi[v];
                closest[(long long)b * npts + pm] = cidx;
                const float* q = pts + ((long long)b * npts + pm) * 3;
                float* acc = accum + ((long long)b * NC + cidx) * 4;
                atomicAdd(acc + 0, q[0]);
                atomicAdd(acc + 1, q[1]);
                atomicAdd(acc + 2, q[2]);
                atomicAdd(acc + 3, 1.0f);
            }
        }
    }
}

// -------------------------------------------------------------------------
// update: centroid = sum/(count+1e-8); refresh prepared B; re-zero accum
// -------------------------------------------------------------------------
__global__ void update_kernel(float* __restrict__ cent,
                              float* __restrict__ centB,
                              float* __restrict__ accum) {
    int i = blockIdx.x * blockDim.x + threadIdx.x;
    if (i < NB * NC) {
        float* a = accum + i * 4;
        float x = a[0], y = a[1], z = a[2], w = a[3];
        float inv = 1.0f / (w + 1e-8f);
        float nx = x * inv, ny = y * inv, nz = z * inv;
        float* c = cent + i * 4;
        c[0] = nx; c[1] = ny; c[2] = nz; c[3] = 0.0f;
        float* cb = centB + i * 4;
        cb[0] = 2.f * nx; cb[1] = 2.f * ny; cb[2] = 2.f * nz;
        cb[3] = -(nx * nx + ny * ny + nz * nz);
        a[0] = 0.f; a[1] = 0.f; a[2] = 0.f; a[3] = 0.f;
    }
}

// -------------------------------------------------------------------------
// topk: one block per (cluster, batch). Scan all points, keep top-15 by full
// sim among points assigned to this cluster; then softmax weights.
// -------------------------------------------------------------------------
__global__ void topk_kernel(const float* __restrict__ pts,
                            const float* __restrict__ cent,
                            const int*  __restrict__ closest,
                            int*   __restrict__ tk_idx,   // (B,NC,15)
                            float* __restrict__ tk_w,     // (B,NC,15)
                            int npts) {
    const int c   = blockIdx.x;
    const int b   = blockIdx.y;
    const int tid = threadIdx.x;

    const float* cp = cent + ((long long)b * NC + c) * 4;
    float cx = cp[0], cy = cp[1], cz = cp[2];
    float cn = cx * cx + cy * cy + cz * cz;

    float lv[MAXN];
    int   li[MAXN];
#pragma unroll
    for (int k = 0; k < MAXN; ++k) { lv[k] = NEGV; li[k] = 0; }

    const float* pb = pts + (long long)b * npts * 3;
    const int*   cl = closest + (long long)b * npts;
    for (int p = tid; p < npts; p += blockDim.x) {
        if (cl[p] == c) {
            float x = pb[p * 3 + 0], y = pb[p * 3 + 1], z = pb[p * 3 + 2];
            float s = 2.f * (x * cx + y * cy + z * cz) - (x * x + y * y + z * z) - cn;
            if (s > lv[MAXN - 1]) {
                float cv = s; int ci = p;
#pragma unroll
                for (int k = 0; k < MAXN; ++k) {
                    if (cv > lv[k]) {
                        float tv = lv[k]; lv[k] = cv; cv = tv;
                        int   ti = li[k]; li[k] = ci; ci = ti;
                    }
                }
            }
        }
    }

    __shared__ float sv[256];
    __shared__ int   si[256];
    __shared__ int   st[256];
    __shared__ float gv[MAXN];
    __shared__ int   gi[MAXN];

    int head = 0;
    for (int r = 0; r < MAXN; ++r) {
        float hv; int hi;
        if (head < MAXN) { hv = lv[head]; hi = li[head]; }
        else             { hv = NEGV;     hi = 0;        }
        sv[tid] = hv; si[tid] = hi; st[tid] = tid;
        __syncthreads();
        for (int s = 128; s > 0; s >>= 1) {
            if (tid < s) {
                if (sv[tid + s] > sv[tid]) {
                    sv[tid] = sv[tid + s]; si[tid] = si[tid + s]; st[tid] = st[tid + s];
                }
            }
            __syncthreads();
        }
        int wt = st[0];
        if (tid == 0) { gv[r] = sv[0]; gi[r] = si[0]; }
        if (tid == wt) head++;
        __syncthreads();
    }

    if (tid == 0) {
        float smax = NEGV;
        for (int r = 0; r < MAXN; ++r)
            if (gv[r] > VALID_THR && gv[r] > smax) smax = gv[r];
        float w[MAXN];
        float sum = 0.f;
        for (int r = 0; r < MAXN; ++r) {
            w[r] = (gv[r] > VALID_THR) ? __expf(gv[r] - smax) : 0.0f;
            sum += w[r];
        }
        float denom = fmaxf(sum, 1e-30f);
        long long o = ((long long)b * NC + c) * MAXN;
        for (int r = 0; r < MAXN; ++r) {
            tk_w[o + r]   = w[r] / denom;
            tk_idx[o + r] = gi[r];
        }
    }
}

// -------------------------------------------------------------------------
// features: out[b][f][c] = sum_k w[k] * ft[b][f][idx[k]]
// -------------------------------------------------------------------------
__global__ void feat_kernel(const float* __restrict__ ft,     // (B,F,npts)
                            const int*   __restrict__ tk_idx,
                            const float* __restrict__ tk_w,
                            float* __restrict__ outf,         // (B,F,NC)
                            int npts, int F) {
    const int c = blockIdx.x;
    const int b = blockIdx.y;
    const int f = threadIdx.x;

    __shared__ int   sidx[MAXN];
    __shared__ float swt[MAXN];
    if (f < MAXN) {
        long long o = ((long long)b * NC + c) * MAXN;
        sidx[f] = tk_idx[o + f];
        swt[f]  = tk_w[o + f];
    }
    __syncthreads();

    if (f < F) {
        const float* fr = ft + ((long long)b * F + f) * npts;
        float acc = 0.f;
#pragma unroll
        for (int k = 0; k < MAXN; ++k) acc += swt[k] * fr[sidx[k]];
        outf[((long long)b * F + f) * NC + c] = acc;
    }
}

// -------------------------------------------------------------------------
// copy final centroids (B,NC,4 padded) -> d_out (B,NC,3)
// -------------------------------------------------------------------------
__global__ void copyout_kernel(const float* __restrict__ cent, float* __restrict__ out) {
    int i = blockIdx.x * blockDim.x + threadIdx.x;
    if (i < NB * NC * 3) {
        int bc = i / 3;
        int d  = i % 3;
        out[i] = cent[bc * 4 + d];
    }
}

extern "C" void kernel_launch(void* const* d_in, const int* in_sizes, int n_in,
                              void* d_out, int out_size, void* d_ws, size_t ws_size,
                              hipStream_t stream) {
    (void)n_in; (void)out_size; (void)ws_size;

    const float* pts = (const float*)d_in[0];   // (4,100000,3)
    const float* ft  = (const float*)d_in[1];   // (4,256,100000)
    const float* cin = (const float*)d_in[2];   // (4,128,3)

    const int npts = in_sizes[0] / (NB * 3);
    const int F    = in_sizes[1] / (NB * npts);

    // workspace carve-up
    float* cent    = (float*)d_ws;                          // NB*NC*4
    float* centB   = cent + NB * NC * 4;                    // NB*NC*4 (prepared B)
    float* accum   = centB + NB * NC * 4;                   // NB*NC*4
    int*   closest = (int*)(accum + NB * NC * 4);           // NB*npts
    int*   tk_idx  = closest + NB * npts;                   // NB*NC*MAXN
    float* tk_w    = (float*)(tk_idx + NB * NC * MAXN);     // NB*NC*MAXN

    float* out_cent = (float*)d_out;            // NB*NC*3
    float* out_feat = out_cent + NB * NC * 3;   // NB*F*NC

    init_kernel<<<(NB * NC + 255) / 256, 256, 0, stream>>>(cin, cent, centB, accum);

    dim3 agrid((npts + 127) / 128, NB);
    for (int it = 0; it < 25; ++it) {
        assign_kernel<<<agrid, 256, 0, stream>>>(pts, centB, closest, accum, npts);
        update_kernel<<<(NB * NC + 255) / 256, 256, 0, stream>>>(cent, centB, accum);
    }

    dim3 cgrid(NC, NB);
    topk_kernel<<<cgrid, 256, 0, stream>>>(pts, cent, closest, tk_idx, tk_w, npts);
    feat_kernel<<<cgrid, 256, 0, stream>>>(ft, tk_idx, tk_w, out_feat, npts, F);
    copyout_kernel<<<(NB * NC * 3 + 255) / 256, 256, 0, stream>>>(cent, out_cent);
}